// NMGaussianMxiture_68350109548955
// MI455X (gfx1250) — compile-verified
//
#include <hip/hip_runtime.h>
#include <math.h>

// ---------------------------------------------------------------------------
// NM Gaussian mixture pairwise-product log-likelihood for gfx1250 (MI455X).
//
// quad[n,p] = (x_n - m_p)^T P_p (x_n - m_p)  is rewritten as <theta_p, phi(x_n)>
// with phi(x) = [x(16), 1, {x_d*x_e, d<=e}(136), pad(7)]  -> K = 160,
// turning the hot loop into a [16384 x 160] @ [160 x 256] f32 GEMM executed
// with chained V_WMMA_F32_16X16X4_F32 (wave32).
// ---------------------------------------------------------------------------

typedef __attribute__((ext_vector_type(2))) float v2f;
typedef __attribute__((ext_vector_type(8))) float v8f;

#define NSAMP  16384
#define KMIX   16
#define DDIM   16
#define NPAIR  256
#define NFEAT  160       // 153 real features padded to 160 (40 chunks of K=4)
#define NCHUNK 40
#define NTILE  (NSAMP / 16)   // 1024 sample tiles
#define WAVES  8              // waves (tiles) per block in the main kernel

// ----- compile-time feature index mapping: f -> (d, e), 16 means "const 1" --
__host__ __device__ constexpr int featD(int f) {
  if (f < 16) return f;      // linear term x_f
  if (f == 16) return 16;    // constant term
  int t = f - 17;
  if (t >= 136) return 16;   // padding
  int d = 0;
  while (t >= 16 - d) { t -= 16 - d; ++d; }
  return d;
}
__host__ __device__ constexpr int featE(int f) {
  if (f < 16) return 16;
  if (f == 16) return 16;
  int t = f - 17;
  if (t >= 136) return 16;
  int d = 0;
  while (t >= 16 - d) { t -= 16 - d; ++d; }
  return d + t;
}

template <int F>
__device__ inline float phiF(const float* x) {
  constexpr int d = featD(F);
  constexpr int e = featE(F);
  const float vd = (d == 16) ? 1.0f : x[d];
  const float ve = (e == 16) ? 1.0f : x[e];
  return vd * ve;
}

// Build per-lane WMMA A-operand chunks: a0[c] = phi[4c+klo], a1[c] = phi[4c+klo+1]
// with klo = 0 (lanes 0-15) or 2 (lanes 16-31), fully unrolled at compile time.
template <int C>
struct BuildA {
  static __device__ inline void run(const float* x, int half, float* a0, float* a1) {
    BuildA<C - 1>::run(x, half, a0, a1);
    const float lo0 = phiF<4 * C + 0>(x);
    const float lo1 = phiF<4 * C + 1>(x);
    const float hi0 = phiF<4 * C + 2>(x);
    const float hi1 = phiF<4 * C + 3>(x);
    a0[C] = half ? hi0 : lo0;
    a1[C] = half ? hi1 : lo1;
  }
};
template <>
struct BuildA<-1> {
  static __device__ inline void run(const float*, int, float*, float*) {}
};

// ----- f64 LU (no pivoting; matrices are SPD) solve + log-determinant -------
__device__ inline void lu_solve16(double M[16][16], double* rhs, double* logdet) {
  double ld = 0.0;
  for (int k = 0; k < 16; ++k) {
    const double piv = M[k][k];
    ld += log(piv);
    const double inv = 1.0 / piv;
    for (int r = k + 1; r < 16; ++r) {
      const double f = M[r][k] * inv;
      M[r][k] = f;
      for (int c = k + 1; c < 16; ++c) M[r][c] -= f * M[k][c];
      rhs[r] -= f * rhs[k];
    }
  }
  for (int k = 15; k >= 0; --k) {
    double s = rhs[k];
    for (int c = k + 1; c < 16; ++c) s -= M[k][c] * rhs[c];
    rhs[k] = s / M[k][k];
  }
  *logdet = ld;
}

// ---------------------------------------------------------------------------
// Kernel 1: setup (1 block x 256 threads).
//  threads 0..15 : A_k = tril(L)tril(L)^T + I ; Sigma_k = inv(A_k) (f64 GJ);
//                  rhs_k = Sigma_k mu_k.
//  all 256      : per pair p=(i,j): P = Sigma_i+Sigma_j, solve P m = r,
//                  logdet P; Asum = A_i+A_j, q = dmu^T inv(Asum) dmu,
//                  logdet Asum; coef_p (log-space); pack theta_p into the
//                  per-lane WMMA B layout (V0 = K klo, V1 = K klo+1).
// ---------------------------------------------------------------------------
__global__ __launch_bounds__(256) void setup_kernel(
    const float* __restrict__ mu, const float* __restrict__ L,
    const float* __restrict__ w,
    double* __restrict__ SigmaW, double* __restrict__ Aw,
    double* __restrict__ rhsW,
    float* __restrict__ Bpack, float* __restrict__ coefW) {
  const int t = threadIdx.x;

  if (t < KMIX) {
    const int k = t;
    double Amat[16][16];
    for (int a = 0; a < 16; ++a)
      for (int b = 0; b < 16; ++b) {
        const int mn = a < b ? a : b;
        double s = (a == b) ? 1.0 : 0.0;
        for (int c = 0; c <= mn; ++c)
          s += (double)L[(k * 16 + a) * 16 + c] * (double)L[(k * 16 + b) * 16 + c];
        Amat[a][b] = s;
      }
    // Gauss-Jordan inverse (SPD -> no pivoting needed)
    double M[16][32];
    for (int a = 0; a < 16; ++a)
      for (int b = 0; b < 16; ++b) {
        M[a][b] = Amat[a][b];
        M[a][16 + b] = (a == b) ? 1.0 : 0.0;
      }
    for (int kk = 0; kk < 16; ++kk) {
      const double inv = 1.0 / M[kk][kk];
      for (int c = 0; c < 32; ++c) M[kk][c] *= inv;
      for (int r = 0; r < 16; ++r) {
        if (r == kk) continue;
        const double f = M[r][kk];
        for (int c = 0; c < 32; ++c) M[r][c] -= f * M[kk][c];
      }
    }
    for (int a = 0; a < 16; ++a)
      for (int b = 0; b < 16; ++b) {
        SigmaW[(k * 16 + a) * 16 + b] = M[a][16 + b];
        Aw[(k * 16 + a) * 16 + b] = Amat[a][b];
      }
    for (int a = 0; a < 16; ++a) {
      double s = 0.0;
      for (int b = 0; b < 16; ++b) s += M[a][16 + b] * (double)mu[k * 16 + b];
      rhsW[k * 16 + a] = s;
    }
  }
  __threadfence();
  __syncthreads();

  // ---- pair phase: one thread per pair ----
  const int p = t;
  const int i = p >> 4, j = p & 15;
  double P[16][16], Work[16][16];
  double r[16], m[16], dmu[16], y[16];

  for (int a = 0; a < 16; ++a)
    for (int b = 0; b < 16; ++b)
      P[a][b] = SigmaW[(i * 16 + a) * 16 + b] + SigmaW[(j * 16 + a) * 16 + b];
  for (int a = 0; a < 16; ++a) {
    r[a] = rhsW[i * 16 + a] + rhsW[j * 16 + a];
    dmu[a] = (double)mu[i * 16 + a] - (double)mu[j * 16 + a];
    m[a] = r[a];
  }

  double ldP;
  for (int a = 0; a < 16; ++a)
    for (int b = 0; b < 16; ++b) Work[a][b] = P[a][b];
  lu_solve16(Work, m, &ldP);  // m = P^{-1} r ; ldP = log det P

  double ldA;
  for (int a = 0; a < 16; ++a)
    for (int b = 0; b < 16; ++b)
      Work[a][b] = Aw[(i * 16 + a) * 16 + b] + Aw[(j * 16 + a) * 16 + b];
  for (int a = 0; a < 16; ++a) y[a] = dmu[a];
  lu_solve16(Work, y, &ldA);  // y = Asum^{-1} dmu ; ldA = log det Asum

  double q = 0.0, cval = 0.0;
  for (int a = 0; a < 16; ++a) {
    q += dmu[a] * y[a];
    cval += r[a] * m[a];  // c = m^T P m = r . m
  }

  const double LOG2PI = 1.8378770664093453;
  // log coef = log w_i + log w_j - q/2 + (log det P)/2 - (log det Asum)/2 - D*log(2pi)
  const double logcoef = log((double)w[i]) + log((double)w[j]) - 0.5 * q +
                         0.5 * ldP - 0.5 * ldA - 16.0 * LOG2PI;
  coefW[p] = (float)exp(logcoef);

  // ---- theta packing into B layout: Bpack[((pt*40+c)*32 + lane)*2 + v] ----
  const int col = p & 15, pt = p >> 4;
  for (int f = 0; f < NFEAT; ++f) {
    double th;
    if (f < 16)
      th = -2.0 * r[f];  // b = P m = r exactly
    else if (f == 16)
      th = cval;
    else if (f < 153) {
      const int d = featD(f), e = featE(f);
      th = (d == e) ? P[d][d] : 2.0 * P[d][e];
    } else
      th = 0.0;
    const int c40 = f >> 2, kk = f & 3;
    const int lane = ((kk >> 1) << 4) + col;  // kk>=2 -> lanes 16..31 (klo=2)
    const int v = kk & 1;
    Bpack[(((pt * NCHUNK) + c40) * 32 + lane) * 2 + v] = (float)th;
  }
}

// ---------------------------------------------------------------------------
// Kernel 2: main GEMM + exp/log. One wave per 16-sample tile.
// ---------------------------------------------------------------------------
__global__ __launch_bounds__(256) void mix_kernel(
    const float* __restrict__ X, const float* __restrict__ Bpack,
    const float* __restrict__ coefW, float* __restrict__ tileW) {
  const int lane = threadIdx.x & 31;
  const int wave = threadIdx.x >> 5;
  const int tile = blockIdx.x * WAVES + wave;
  const int row = lane & 15;   // sample row (A layout M) and pair column (C layout N)
  const int half = lane >> 4;  // selects K pair {0,1} vs {2,3} within each chunk

  // Load this lane's sample row (rows duplicated across the two lane halves).
  const float* xr = X + (tile * 16 + row) * 16;
  float x[16];
#pragma unroll
  for (int iq = 0; iq < 16; ++iq) x[iq] = xr[iq];

  // Build all 40 A-operand chunks (phi features) once; reused for 16 ptiles.
  float a0[NCHUNK], a1[NCHUNK];
  BuildA<NCHUNK - 1>::run(x, half, a0, a1);

  v8f partial = {0.f, 0.f, 0.f, 0.f, 0.f, 0.f, 0.f, 0.f};
  const float2* bp = (const float2*)Bpack;
  const float cf = coefW[lane & 15];  // placeholder read pattern; real read below

#pragma unroll 1
  for (int pt = 0; pt < 16; ++pt) {
    v8f acc = {0.f, 0.f, 0.f, 0.f, 0.f, 0.f, 0.f, 0.f};
    const float2* bpp = bp + (size_t)(pt * NCHUNK) * 32 + lane;
#pragma unroll
    for (int c = 0; c < NCHUNK; ++c) {
      const float2 b = bpp[c * 32];
      v2f av;
      av[0] = a0[c];
      av[1] = a1[c];
      v2f bv;
      bv[0] = b.x;
      bv[1] = b.y;
      acc = __builtin_amdgcn_wmma_f32_16x16x4_f32(
          false, av, false, bv, (short)0, acc, false, false);
    }
    // C layout: vgpr g, this lane -> quad[sample row g+8*half][pair pt*16+row]
    const float cfp = coefW[pt * 16 + row];
#pragma unroll
    for (int g = 0; g < 8; ++g)
      partial[g] += cfp * __expf(-0.5f * acc[g]);
  }
  (void)cf;

  // Reduce over the 16 pair-columns held across each 16-lane half.
  float lsum = 0.0f;
#pragma unroll
  for (int g = 0; g < 8; ++g) {
    float v = partial[g];
    v += __shfl_xor(v, 1, 32);
    v += __shfl_xor(v, 2, 32);
    v += __shfl_xor(v, 4, 32);
    v += __shfl_xor(v, 8, 32);
    lsum += __logf(v);  // log pdf of sample row g + 8*half
  }
  lsum += __shfl_xor(lsum, 16, 32);  // combine the two halves (rows 0-7 + 8-15)
  if (lane == 0) tileW[tile] = lsum;
}

// ---------------------------------------------------------------------------
// Kernel 3: deterministic tree reduction of the 1024 per-tile log sums.
// ---------------------------------------------------------------------------
__global__ __launch_bounds__(256) void reduce_kernel(
    const float* __restrict__ tileW, float* __restrict__ out) {
  __shared__ float s[256];
  const int t = threadIdx.x;
  float v = tileW[t] + tileW[t + 256] + tileW[t + 512] + tileW[t + 768];
  s[t] = v;
  __syncthreads();
  for (int off = 128; off > 0; off >>= 1) {
    if (t < off) s[t] += s[t + off];
    __syncthreads();
  }
  if (t == 0) out[0] = s[0] / (float)NSAMP;
}

// ---------------------------------------------------------------------------
// Launch wrapper. ws usage ~232 KB:
//   SigmaW (f64 4096) | Aw (f64 4096) | rhsW (f64 256) |
//   Bpack (f32 40960) | coefW (f32 256) | tileW (f32 1024)
// ---------------------------------------------------------------------------
extern "C" void kernel_launch(void* const* d_in, const int* in_sizes, int n_in,
                              void* d_out, int out_size, void* d_ws,
                              size_t ws_size, hipStream_t stream) {
  const float* X = (const float*)d_in[0];
  const float* mu = (const float*)d_in[1];
  const float* L = (const float*)d_in[2];
  const float* w = (const float*)d_in[3];
  // d_in[4] = it (unused by the reference math)

  double* SigmaW = (double*)d_ws;
  double* Aw = SigmaW + 16 * 16 * 16;
  double* rhsW = Aw + 16 * 16 * 16;
  float* Bpack = (float*)(rhsW + 16 * 16);
  float* coefW = Bpack + 16 * NCHUNK * 32 * 2;
  float* tileW = coefW + NPAIR;
  float* out = (float*)d_out;

  setup_kernel<<<1, 256, 0, stream>>>(mu, L, w, SigmaW, Aw, rhsW, Bpack, coefW);
  mix_kernel<<<NTILE / WAVES, 256, 0, stream>>>(X, Bpack, coefW, tileW);
  reduce_kernel<<<1, 256, 0, stream>>>(tileW, out);
}